// SetCriterion_21869973471695
// MI455X (gfx1250) — compile-verified
//
#include <hip/hip_runtime.h>
#include <math.h>

// Problem constants (match reference)
#define NUM_CL     14         // NUM_CLASSES + 1
#define NO_OBJ     13
#define ROWS       1600000    // B*T*N = 32*1000*50
#define TILE_ROWS  256
#define NTILES     (ROWS / TILE_ROWS)      // 6250
#define K1_BLOCKS  625                     // 10 tiles per block, exact
#define TILE_FLOATS (TILE_ROWS * NUM_CL)   // 3584 floats = 14336 B
#define TILE_CHUNKS ((TILE_FLOATS * 4) / 16) // 896 x 16B chunks

#define GLOBAL_AS __attribute__((address_space(1)))
#define LDS_AS    __attribute__((address_space(3)))

// 16-byte generic vector type matching the builtin's expected pointee
// (diagnostic showed: __attribute__((__vector_size__(4 * sizeof(int)))) int)
typedef int async_v4i __attribute__((vector_size(4 * sizeof(int))));

#if defined(__HIP_DEVICE_COMPILE__)
#if __has_builtin(__builtin_amdgcn_global_load_async_to_lds_b128)
#define HAVE_ASYNC_LDS 1
#endif
#endif

// Stage one 256-row x 14-class logit tile into LDS.
// Preferred path: gfx1250 async DMA (GLOBAL_LOAD_ASYNC_TO_LDS_B128, tracked by ASYNCcnt).
__device__ __forceinline__ void tile_copy(const float* __restrict__ g, float* l) {
#ifdef HAVE_ASYNC_LDS
  for (int i = threadIdx.x; i < TILE_CHUNKS; i += TILE_ROWS) {
    __builtin_amdgcn_global_load_async_to_lds_b128(
        (GLOBAL_AS async_v4i*)(g + i * 4),
        (LDS_AS async_v4i*)(l + i * 4),
        /*offset=*/0, /*cpol=*/0);
  }
#else
  for (int i = threadIdx.x; i < TILE_CHUNKS; i += TILE_ROWS) {
    reinterpret_cast<float4*>(l)[i] = reinterpret_cast<const float4*>(g)[i];
  }
#endif
}

__device__ __forceinline__ void wait_async_zero() {
#if __has_builtin(__builtin_amdgcn_s_wait_asynccnt)
  __builtin_amdgcn_s_wait_asynccnt(0);
#else
  asm volatile("s_wait_asynccnt 0" ::: "memory");
#endif
}

__global__ __launch_bounds__(TILE_ROWS) void setcrit_partials(
    const float* __restrict__ logits,   // [ROWS, 14]
    const float* __restrict__ pdoa,     // [ROWS, 3]
    const float* __restrict__ tdoa,     // [ROWS, 3]
    const float* __restrict__ eweight,  // [14]
    const int*   __restrict__ tcls,     // [ROWS]
    float* __restrict__ partials)       // [gridDim.x, 3]
{
  __shared__ float buf[2][TILE_FLOATS];     // double-buffered logit tiles (28 KB)
  __shared__ float red0[TILE_ROWS];
  __shared__ float red1[TILE_ROWS];
  __shared__ float red2[TILE_ROWS];

  const int tid = threadIdx.x;
  float ce_acc = 0.f, doa_acc = 0.f, cnt_acc = 0.f;

  int tile = blockIdx.x;
  int cur = 0;
  if (tile < NTILES)
    tile_copy(logits + (long)tile * TILE_FLOATS, buf[0]);

  for (; tile < NTILES; tile += gridDim.x) {
    const int nxt = tile + gridDim.x;
    wait_async_zero();      // this wave's async loads into buf[cur] are complete
    __syncthreads();        // all waves' loads complete; everyone done with buf[cur^1]
    if (nxt < NTILES)       // kick off DMA for the next tile while we compute
      tile_copy(logits + (long)nxt * TILE_FLOATS, buf[cur ^ 1]);

    const int row = tile * TILE_ROWS + tid;
    if (nxt < NTILES) {     // gfx1250 global_prefetch_b8 for next tile's side streams
      const int nrow = nxt * TILE_ROWS + tid;
      __builtin_prefetch(&tcls[nrow], 0, 1);
      __builtin_prefetch(&pdoa[(long)nrow * 3], 0, 1);
      __builtin_prefetch(&tdoa[(long)nrow * 3], 0, 1);
    }

    // ---- weighted CE for this row ----
    const float* rl = &buf[cur][tid * NUM_CL];
    float m = rl[0];
#pragma unroll
    for (int c = 1; c < NUM_CL; ++c) m = fmaxf(m, rl[c]);
    float s = 0.f;
#pragma unroll
    for (int c = 0; c < NUM_CL; ++c) s += expf(rl[c] - m);
    const float lse = m + logf(s);
    const int   t   = tcls[row];
    ce_acc += eweight[t] * (lse - rl[t]);

    // ---- masked L1 DOA ----
    if (t != NO_OBJ) {
      const long b = (long)row * 3;
      doa_acc += fabsf(pdoa[b + 0] - tdoa[b + 0])
               + fabsf(pdoa[b + 1] - tdoa[b + 1])
               + fabsf(pdoa[b + 2] - tdoa[b + 2]);
      cnt_acc += 1.0f;
    }
    cur ^= 1;
  }

  // deterministic block tree reduction
  red0[tid] = ce_acc; red1[tid] = doa_acc; red2[tid] = cnt_acc;
  __syncthreads();
  for (int off = TILE_ROWS / 2; off > 0; off >>= 1) {
    if (tid < off) {
      red0[tid] += red0[tid + off];
      red1[tid] += red1[tid + off];
      red2[tid] += red2[tid + off];
    }
    __syncthreads();
  }
  if (tid == 0) {
    partials[blockIdx.x * 3 + 0] = red0[0];
    partials[blockIdx.x * 3 + 1] = red1[0];
    partials[blockIdx.x * 3 + 2] = red2[0];
  }
}

__global__ __launch_bounds__(256) void setcrit_final(
    const float* __restrict__ partials, float* __restrict__ out)
{
  __shared__ float r0[256], r1[256], r2[256];
  float a = 0.f, b = 0.f, c = 0.f;
  for (int i = threadIdx.x; i < K1_BLOCKS; i += 256) {
    a += partials[3 * i + 0];
    b += partials[3 * i + 1];
    c += partials[3 * i + 2];
  }
  r0[threadIdx.x] = a; r1[threadIdx.x] = b; r2[threadIdx.x] = c;
  __syncthreads();
  for (int off = 128; off > 0; off >>= 1) {
    if (threadIdx.x < off) {
      r0[threadIdx.x] += r0[threadIdx.x + off];
      r1[threadIdx.x] += r1[threadIdx.x + off];
      r2[threadIdx.x] += r2[threadIdx.x + off];
    }
    __syncthreads();
  }
  if (threadIdx.x == 0) {
    const float loss_class = r0[0] / (float)ROWS;         // mean over all rows
    const float n_elems    = r2[0] * 3.0f;
    const float loss_doa   = (n_elems > 0.f) ? (r1[0] / fmaxf(n_elems, 1.0f)) : 0.0f;
    out[0] = 1.0f * loss_class + 2.0f * loss_doa;         // W_CLASS, W_DOA
  }
}

extern "C" void kernel_launch(void* const* d_in, const int* in_sizes, int n_in,
                              void* d_out, int out_size, void* d_ws, size_t ws_size,
                              hipStream_t stream) {
  (void)in_sizes; (void)n_in; (void)out_size; (void)ws_size;
  const float* logits = (const float*)d_in[0];  // pred_logits [32,1000,50,14]
  const float* pdoa   = (const float*)d_in[1];  // pred_doa    [32,1000,50,3]
  const float* tdoa   = (const float*)d_in[2];  // target_doa  [32000,50,3]
  const float* ew     = (const float*)d_in[3];  // empty_weight [14]
  const int*   tcls   = (const int*)  d_in[4];  // target_classes [32000,50]
  float* partials = (float*)d_ws;               // 625*3 floats

  setcrit_partials<<<K1_BLOCKS, TILE_ROWS, 0, stream>>>(logits, pdoa, tdoa, ew, tcls, partials);
  setcrit_final<<<1, 256, 0, stream>>>(partials, (float*)d_out);
}